// GroupBasis_73796128080153
// MI455X (gfx1250) — compile-verified
//
#include <hip/hip_runtime.h>

typedef __attribute__((ext_vector_type(2))) float v2f;
typedef __attribute__((ext_vector_type(4))) float v4f;
typedef __attribute__((ext_vector_type(8))) float v8f;

#define NUM_BASIS 16
#define BS 128
#define CHARTS 256
#define REG_FAC 1.0f

// Full 16x16 f32 matmul: C = X * Y, X and Y row-major 16x16 in LDS.
// Uses 4 chained V_WMMA_F32_16X16X4_F32 (K-slices of 4).
// A-layout (ISA 7.12.2, 32-bit A 16x4): lanes 0-15 hold (M=lane, K=k0,k0+1),
// lanes 16-31 hold (M=lane-16, K=k0+2,k0+3). B rows striped across lanes.
__device__ __forceinline__ v8f mat16_mul(const float* __restrict__ X,
                                         const float* __restrict__ Y,
                                         int lane) {
  v8f acc = {};
  const int half = (lane >= 16) ? 2 : 0;
  const int m = lane & 15;
#pragma unroll
  for (int k0 = 0; k0 < 16; k0 += 4) {
    const int ka = k0 + half;
    v2f a;
    a.x = X[m * 16 + ka];
    a.y = X[m * 16 + ka + 1];
    v2f bb;
    bb.x = Y[ka * 16 + m];
    bb.y = Y[(ka + 1) * 16 + m];
    acc = __builtin_amdgcn_wmma_f32_16x16x4_f32(
        /*neg_a=*/false, a, /*neg_b=*/false, bb,
        /*c_mod=*/(short)0, acc, /*reuse_a=*/false, /*reuse_b=*/false);
  }
  return acc;
}

// One wave32 computes one expm(16x16). Scaling-and-squaring (s=4) with
// order-8 Taylor via Horner; all matmuls on the WMMA pipe.
__global__ __launch_bounds__(256) void expm_kernel(
    const float* __restrict__ coeffs,     // [128,256,16]
    const float* __restrict__ lie_basis,  // [16,16,16]
    const float* __restrict__ in_basis,   // [16,16,16]
    const float* __restrict__ out_basis,  // [16,16,16]
    float* __restrict__ out)              // [3,128,256,16,16]
{
  __shared__ __align__(16) float sb[NUM_BASIS * 256];  // basis set, row-major
  __shared__ __align__(16) float bufB[8][256];         // per-wave B = A/16
  __shared__ __align__(16) float bufT[8][256];         // per-wave running T
  __shared__ float invf[NUM_BASIS];

  const int t = threadIdx.x;
  const int wave = t >> 5;
  const int lane = t & 31;
  const int type = blockIdx.x >> 12;          // 4096 blocks per basis type
  const int grp = blockIdx.x & 4095;
  const int flat = grp * 8 + wave;            // (b,c) index in [0, 32768)
  const int b = flat >> 8;
  const int c = flat & 255;

  const float* basis = (type == 0) ? lie_basis : (type == 1) ? in_basis : out_basis;

  // Stage the 16 basis matrices into LDS (coalesced).
#pragma unroll
  for (int i = 0; i < NUM_BASIS; ++i)
    sb[i * 256 + t] = basis[i * 256 + t];
  __syncthreads();

  // Lie basis normalization factors: x / sqrt(sum(x^2)/16). Fold into coeffs.
  if (t < NUM_BASIS) {
    if (type == 0) {
      float s = 0.f;
      for (int i = 0; i < 256; ++i) {
        float x = sb[t * 256 + i];
        s += x * x;
      }
      invf[t] = rsqrtf(s * (1.0f / 16.0f));
    } else {
      invf[t] = 1.0f;
    }
  }
  __syncthreads();

  // Per-wave scaled coefficients: c_k * invf_k * 2^-4 (scaling step of expm).
  const float* cp = coeffs + (size_t)(b * CHARTS + c) * NUM_BASIS;
  __builtin_prefetch(cp, 0, 0);
  float cc[NUM_BASIS];
#pragma unroll
  for (int k = 0; k < NUM_BASIS; ++k) cc[k] = cp[k] * invf[k] * 0.0625f;

  // Build B = sum_k cc_k * basis_k directly in WMMA C/D layout:
  // lane<16 VGPR v holds (M=v,   N=lane); lane>=16 holds (M=v+8, N=lane-16).
  const int lofs = (lane < 16) ? lane : (lane + 112);  // v*16+lofs == m*16+n
  float bm[8];
#pragma unroll
  for (int v = 0; v < 8; ++v) bm[v] = 0.f;
  for (int k = 0; k < NUM_BASIS; ++k) {
    const float ck = cc[k];
#pragma unroll
    for (int v = 0; v < 8; ++v) bm[v] += ck * sb[k * 256 + v * 16 + lofs];
  }

  float* __restrict__ Bb = bufB[wave];
  float* __restrict__ Tb = bufT[wave];
#pragma unroll
  for (int v = 0; v < 8; ++v) Bb[v * 16 + lofs] = bm[v];

  // T = I + B/8  (innermost Horner term, done in VALU).
#pragma unroll
  for (int v = 0; v < 8; ++v) {
    const float d = ((lane == v) || (lane == v + 24)) ? 1.0f : 0.0f;
    Tb[v * 16 + lofs] = bm[v] * 0.125f + d;
  }

  // Horner: T <- (B*T)/j + I for j = 7..1  => T = sum_{i<=8} B^i / i!
  for (int j = 7; j >= 1; --j) {
    v8f p = mat16_mul(Bb, Tb, lane);
    const float invj = 1.0f / (float)j;
#pragma unroll
    for (int v = 0; v < 8; ++v) {
      const float d = ((lane == v) || (lane == v + 24)) ? 1.0f : 0.0f;
      Tb[v * 16 + lofs] = p[v] * invj + d;
    }
  }

  // Squaring phase: T <- T*T, four times (undo the 2^-4 scaling).
  for (int s = 0; s < 4; ++s) {
    v8f p = mat16_mul(Tb, Tb, lane);
#pragma unroll
    for (int v = 0; v < 8; ++v) Tb[v * 16 + lofs] = p[v];
  }

  // Coalesced output: each lane writes 8 contiguous floats (2x b128).
  float* __restrict__ o =
      out + ((size_t)((type * BS + b) * CHARTS + c)) * 256;
  v4f r0 = *(const v4f*)&Tb[lane * 8];
  v4f r1 = *(const v4f*)&Tb[lane * 8 + 4];
  *(v4f*)&o[lane * 8] = r0;
  *(v4f*)&o[lane * 8 + 4] = r1;
}

// Scalar regularizer: REG_FAC * sum_{b<c} |gram(|x_b|,|x_c|)| on the
// normalized lie basis. One block, LDS tree reduction.
__global__ __launch_bounds__(256) void reg_kernel(
    const float* __restrict__ lie, float* __restrict__ out_reg) {
  __shared__ float sb[NUM_BASIS * 256];
  __shared__ float fr[NUM_BASIS];
  __shared__ float red[256];

  const int t = threadIdx.x;
#pragma unroll
  for (int i = 0; i < NUM_BASIS; ++i) sb[i * 256 + t] = lie[i * 256 + t];
  __syncthreads();

  if (t < NUM_BASIS) {
    float s = 0.f;
    for (int i = 0; i < 256; ++i) {
      float x = sb[t * 256 + i];
      s += x * x;
    }
    fr[t] = rsqrtf(s * (1.0f / 16.0f));
  }
  __syncthreads();

  const int bi = t >> 4;
  const int ci = t & 15;
  float g = 0.f;
  if (bi < ci) {  // strict upper triangle
    float s = 0.f;
    for (int i = 0; i < 256; ++i)
      s += fabsf(sb[bi * 256 + i]) * fabsf(sb[ci * 256 + i]);
    g = fabsf(s * fr[bi] * fr[ci]);
  }
  red[t] = g;
  __syncthreads();
  for (int off = 128; off > 0; off >>= 1) {
    if (t < off) red[t] += red[t + off];
    __syncthreads();
  }
  if (t == 0) *out_reg = REG_FAC * red[0];
}

extern "C" void kernel_launch(void* const* d_in, const int* in_sizes, int n_in,
                              void* d_out, int out_size, void* d_ws,
                              size_t ws_size, hipStream_t stream) {
  const float* coeffs = (const float*)d_in[0];
  const float* lie = (const float*)d_in[1];
  const float* inb = (const float*)d_in[2];
  const float* outb = (const float*)d_in[3];
  float* out = (float*)d_out;

  // 3 types * (128*256 matrices) / (8 waves per block) = 12288 blocks
  expm_kernel<<<dim3(3 * 4096), dim3(256), 0, stream>>>(coeffs, lie, inb, outb,
                                                        out);
  reg_kernel<<<dim3(1), dim3(256), 0, stream>>>(
      lie, out + (size_t)3 * BS * CHARTS * 256);
}